// LovaszSoftmaxLoss_88991722373275
// MI455X (gfx1250) — compile-verified
//
#include <hip/hip_runtime.h>
#include <hip/hip_bf16.h>

// ---------------------------------------------------------------------------
// Lovasz-Softmax loss, MI455X (gfx1250, wave32).
// Pipeline: keygen(softmax+|fg-p|, pack fg into key LSB)
//           -> 4x LSD radix-8 sort per class (hist, block-scan, WMMA digit-
//              scan, ballot-stable scatter)
//           -> cumfg scan + per-element Jaccard-delta contribution
//           -> deterministic tree reductions.
// v_wmma_f32_16x16x4_f32 performs the 256-bin digit prefix sums (X*U and
// Lstrict*rowTotals, 8 WMMAs per class per pass), exact in f32 (counts<2^24).
// ---------------------------------------------------------------------------

#define NDIM   8
#define CDIM   19
#define HWDIM  (512*512)            // 1<<18
#define PDIM   (NDIM*HWDIM)         // 2,097,152
#define TILE   4096
#define NBLK   (PDIM/TILE)          // 512

typedef float v8f __attribute__((ext_vector_type(8)));
typedef float v2f __attribute__((ext_vector_type(2)));

__device__ __forceinline__ unsigned lv_wave_incl_scan(unsigned v) {
  int lane = threadIdx.x & 31;
  #pragma unroll
  for (int d = 1; d < 32; d <<= 1) {
    unsigned n = __shfl_up(v, d, 32);
    if (lane >= d) v += n;
  }
  return v;
}

// ---------------- K0: zero fg totals (ws is poisoned with 0xAA) ------------
__global__ void lv_zero_fg(unsigned* fgTotU) {
  if (threadIdx.x < CDIM) fgTotU[threadIdx.x] = 0u;
}

// ---------------- K5: per-class foreground pixel counts --------------------
__global__ void lv_fg_count(const int* __restrict__ labels, unsigned* fgTotU) {
  __shared__ unsigned h[32];
  if (threadIdx.x < 32) h[threadIdx.x] = 0u;
  __syncthreads();
  int base = blockIdx.x * TILE + threadIdx.x;
  #pragma unroll
  for (int i = 0; i < 16; ++i) {
    int lb = labels[base + i * 256];
    atomicAdd(&h[lb], 1u);
  }
  __syncthreads();
  if (threadIdx.x < CDIM && h[threadIdx.x]) atomicAdd(&fgTotU[threadIdx.x], h[threadIdx.x]);
}

// ---------------- K1: softmax + error keys ---------------------------------
// key = (float_bits(err) << 1) | fg   (err in [0,1] -> bits fit in 31 bits)
__global__ void lv_keygen(const float* __restrict__ logits,
                          const int* __restrict__ labels,
                          unsigned* __restrict__ keys, int c0, int gcnt) {
  int p  = blockIdx.x * 256 + threadIdx.x;
  int n  = p >> 18;
  int hw = p & (HWDIM - 1);
  const float* lg = logits + (size_t)n * CDIM * HWDIM + hw;
  float v[CDIM];
  float mx = -1e30f;
  #pragma unroll
  for (int c = 0; c < CDIM; ++c) { v[c] = lg[(size_t)c * HWDIM]; mx = fmaxf(mx, v[c]); }
  float s = 0.f;
  #pragma unroll
  for (int c = 0; c < CDIM; ++c) { v[c] = __expf(v[c] - mx); s += v[c]; }
  float inv = 1.f / s;
  int lb = labels[p];
  #pragma unroll
  for (int c = 0; c < CDIM; ++c) {
    if (c >= c0 && c < c0 + gcnt) {
      float    pr  = v[c] * inv;
      unsigned fg  = (lb == c) ? 1u : 0u;
      float    err = fabsf((float)fg - pr);
      keys[(size_t)(c - c0) * PDIM + p] = (__float_as_uint(err) << 1) | fg;
    }
  }
}

// ---------------- K2: per-block digit histogram (sort pass) ----------------
__global__ void lv_hist_pass(const unsigned* __restrict__ keys,
                             unsigned* __restrict__ blockHist, int shift) {
  int g = blockIdx.y, b = blockIdx.x;
  __shared__ unsigned h[256];
  h[threadIdx.x] = 0u;
  __syncthreads();
  const unsigned* k = keys + (size_t)g * PDIM + (size_t)b * TILE;
  #pragma unroll
  for (int i = 0; i < 16; ++i) {
    unsigned kk = ~k[threadIdx.x + 256 * i];          // ascending ~key == descending key
    atomicAdd(&h[(kk >> shift) & 255u], 1u);
  }
  __syncthreads();
  blockHist[(size_t)(g * 256 + threadIdx.x) * NBLK + b] = h[threadIdx.x];
}

// ---------------- K3a: exclusive scan over blocks per (class,digit) --------
__global__ void lv_scan_block_hist(unsigned* __restrict__ blockHist,
                                   unsigned* __restrict__ digitTotal, int gcnt) {
  int lane = threadIdx.x & 31;
  int wid  = blockIdx.x * (blockDim.x >> 5) + (threadIdx.x >> 5);
  if (wid >= gcnt * 256) return;
  unsigned* bh = blockHist + (size_t)wid * NBLK;
  unsigned carry = 0u;
  for (int b0 = 0; b0 < NBLK; b0 += 32) {
    unsigned v   = bh[b0 + lane];
    unsigned inc = lv_wave_incl_scan(v);
    bh[b0 + lane] = carry + inc - v;
    carry += __shfl(inc, 31, 32);
  }
  if (lane == 0) digitTotal[wid] = carry;
}

// ---------------- K3b: 256-bin exclusive digit scan via WMMA ---------------
// inclusive_scan(x)[16m+n] = (X*U)[m][n] + (Lstrict*rowTot)[m][n]
// A 16x4 layout: lane<16 -> row=lane, {K0,K1}; lane>=16 -> row=lane-16, {K2,K3}
// B 4x16 / D 16x16: col = lane&15; D vgpr v -> rows v (lo lanes) / v+8 (hi).
__global__ void lv_digit_scan_wmma(const unsigned* __restrict__ digitTotal,
                                   unsigned* __restrict__ digitBase) {
  __shared__ float xs[256];
  __shared__ float rt[16];
  int g = blockIdx.x;
  int l = threadIdx.x;                       // 0..31 (one wave)
  const unsigned* dt = digitTotal + (size_t)g * 256;
  #pragma unroll
  for (int i = 0; i < 8; ++i) xs[l + 32 * i] = (float)dt[l + 32 * i];
  __syncthreads();

  bool hi = (l >= 16);
  int  m  = l & 15;                          // A: row  |  B/D: column
  v8f acc = {};
  #pragma unroll
  for (int t = 0; t < 4; ++t) {              // acc = X * U  (upper-tri ones)
    int k0 = 4 * t + (hi ? 2 : 0);
    int k1 = k0 + 1;
    v2f a, b;
    a.x = xs[16 * m + k0];
    a.y = xs[16 * m + k1];
    b.x = (k0 <= m) ? 1.0f : 0.0f;
    b.y = (k1 <= m) ? 1.0f : 0.0f;
    acc = __builtin_amdgcn_wmma_f32_16x16x4_f32(false, a, false, b, (short)0,
                                                acc, false, false);
  }
  if (m == 15) {                             // column 15 = row totals
    #pragma unroll
    for (int v = 0; v < 8; ++v) rt[(hi ? 8 : 0) + v] = acc[v];
  }
  __syncthreads();
  #pragma unroll
  for (int t = 0; t < 4; ++t) {              // acc += Lstrict * rowTot
    int k0 = 4 * t + (hi ? 2 : 0);
    int k1 = k0 + 1;
    v2f a, b;
    a.x = (k0 < m) ? 1.0f : 0.0f;
    a.y = (k1 < m) ? 1.0f : 0.0f;
    b.x = rt[k0];
    b.y = rt[k1];
    acc = __builtin_amdgcn_wmma_f32_16x16x4_f32(false, a, false, b, (short)0,
                                                acc, false, false);
  }
  unsigned* db = digitBase + (size_t)g * 256;
  #pragma unroll
  for (int v = 0; v < 8; ++v) {
    int row = (hi ? 8 : 0) + v;
    int i   = 16 * row + m;
    float excl = acc[v] - xs[i];             // exclusive = inclusive - x
    db[i] = (unsigned)(excl + 0.5f);
  }
}

// ---------------- K4: stable radix scatter (wave32 ballot matching) --------
__global__ void lv_scatter_pass(const unsigned* __restrict__ keysIn,
                                unsigned* __restrict__ keysOut,
                                const unsigned* __restrict__ blockHistExcl,
                                const unsigned* __restrict__ digitBase, int shift) {
  int g = blockIdx.y, b = blockIdx.x;
  int t = threadIdx.x, lane = t & 31, w = t >> 5;
  __shared__ unsigned waveOff[8 * 256];
  #pragma unroll
  for (int i = 0; i < 8; ++i) waveOff[t + 256 * i] = 0u;
  __syncthreads();

  const unsigned* kin = keysIn + (size_t)g * PDIM + (size_t)b * TILE + (size_t)w * 512;
  unsigned keyR[16], posR[16];
  unsigned lmlt = (1u << lane) - 1u;
  #pragma unroll
  for (int j = 0; j < 16; ++j) {             // rounds preserve memory order
    unsigned key = kin[j * 32 + lane];
    unsigned d   = ((~key) >> shift) & 255u;
    unsigned mask = 0xffffffffu;
    #pragma unroll
    for (int bit = 0; bit < 8; ++bit) {
      unsigned bal = (unsigned)__builtin_amdgcn_ballot_w32(((d >> bit) & 1u) != 0u);
      mask &= ((d >> bit) & 1u) ? bal : ~bal;
    }
    unsigned rank  = __popc(mask & lmlt);
    unsigned grp   = __popc(mask);
    unsigned basew = waveOff[w * 256 + d];   // in-wave DS ops are in order
    if (rank == 0) waveOff[w * 256 + d] = basew + grp;
    keyR[j] = key;
    posR[j] = basew + rank;
  }
  __syncthreads();
  {                                          // thread t handles digit t
    unsigned ex = digitBase[g * 256 + t] +
                  blockHistExcl[(size_t)(g * 256 + t) * NBLK + b];
    #pragma unroll
    for (int ww = 0; ww < 8; ++ww) {
      unsigned c = waveOff[ww * 256 + t];
      waveOff[ww * 256 + t] = ex;
      ex += c;
    }
  }
  __syncthreads();
  unsigned* kout = keysOut + (size_t)g * PDIM;
  #pragma unroll
  for (int j = 0; j < 16; ++j) {
    unsigned d = ((~keyR[j]) >> shift) & 255u;
    kout[waveOff[w * 256 + d] + posR[j]] = keyR[j];
  }
}

// ---------------- K6a: fg counts per sorted chunk --------------------------
__global__ void lv_fg_chunk(const unsigned* __restrict__ keys,
                            unsigned* __restrict__ chunkFg) {
  int g = blockIdx.y, b = blockIdx.x, t = threadIdx.x, lane = t & 31, w = t >> 5;
  const unsigned* k = keys + (size_t)g * PDIM + (size_t)b * TILE;
  unsigned c = 0;
  #pragma unroll
  for (int i = 0; i < 16; ++i) c += k[t + 256 * i] & 1u;
  #pragma unroll
  for (int d = 16; d; d >>= 1) c += __shfl_down(c, d, 32);
  __shared__ unsigned ws[8];
  if (lane == 0) ws[w] = c;
  __syncthreads();
  if (t == 0) {
    unsigned s = 0;
    #pragma unroll
    for (int ww = 0; ww < 8; ++ww) s += ws[ww];
    chunkFg[(size_t)g * NBLK + b] = s;
  }
}

// ---------------- K6b: exclusive scan of chunk fg counts -------------------
__global__ void lv_scan_chunkfg(unsigned* __restrict__ chunkFg) {
  int g = blockIdx.x, lane = threadIdx.x & 31;
  unsigned* p = chunkFg + (size_t)g * NBLK;
  unsigned carry = 0u;
  for (int b0 = 0; b0 < NBLK; b0 += 32) {
    unsigned v   = p[b0 + lane];
    unsigned inc = lv_wave_incl_scan(v);
    p[b0 + lane] = carry + inc - v;
    carry += __shfl(inc, 31, 32);
  }
}

__device__ __forceinline__ float lv_jac(float G, float k, float cf) {
  float den = G + k - cf;
  return (den > 0.f) ? (1.f - (G - cf) / den) : 0.f;
}

// ---------------- K6c: per-element Lovasz contribution ---------------------
__global__ void lv_contrib(const unsigned* __restrict__ keys,
                           const unsigned* __restrict__ chunkFgExcl,
                           const unsigned* __restrict__ fgTotU,
                           float* __restrict__ partials, int c0) {
  int g = blockIdx.y, b = blockIdx.x, t = threadIdx.x, lane = t & 31, w = t >> 5;
  const unsigned* k = keys + (size_t)g * PDIM + (size_t)b * TILE + (size_t)t * 16;
  unsigned keyR[16], lp[16], run = 0;
  #pragma unroll
  for (int j = 0; j < 16; ++j) { keyR[j] = k[j]; run += keyR[j] & 1u; lp[j] = run; }
  unsigned tsum  = run;
  unsigned tIncl = lv_wave_incl_scan(tsum);
  __shared__ unsigned wsums[8];
  __shared__ float    fred[8];
  if (lane == 31) wsums[w] = tIncl;
  __syncthreads();
  unsigned wOff = 0;
  for (int ww = 0; ww < w; ++ww) wOff += wsums[ww];
  unsigned base = chunkFgExcl[(size_t)g * NBLK + b] + wOff + tIncl - tsum;
  float  Gf   = (float)fgTotU[c0 + g];
  size_t idx0 = (size_t)b * TILE + (size_t)t * 16;
  float  acc  = 0.f;
  #pragma unroll
  for (int j = 0; j < 16; ++j) {
    unsigned key = keyR[j];
    float e   = __uint_as_float(key >> 1);
    float cf  = (float)(base + lp[j]);       // inclusive cumfg at this element
    float fgi = (float)(key & 1u);
    float k1  = (float)(idx0 + j + 1);
    float Jc  = lv_jac(Gf, k1, cf);
    float Jp  = lv_jac(Gf, k1 - 1.f, cf - fgi);
    acc += e * (Jc - Jp);
  }
  #pragma unroll
  for (int d = 16; d; d >>= 1) acc += __shfl_down(acc, d, 32);
  if (lane == 0) fred[w] = acc;
  __syncthreads();
  if (t == 0) {
    float s = 0.f;
    #pragma unroll
    for (int ww = 0; ww < 8; ++ww) s += fred[ww];
    partials[(size_t)g * NBLK + b] = s;      // deterministic (no float atomics)
  }
}

// ---------------- K7: fixed-order per-class reduction ----------------------
__global__ void lv_class_reduce(const float* __restrict__ partials,
                                float* __restrict__ classLoss, int c0) {
  int g = blockIdx.x, t = threadIdx.x;
  __shared__ float s[256];
  s[t] = partials[(size_t)g * NBLK + t] + partials[(size_t)g * NBLK + t + 256];
  __syncthreads();
  for (int st = 128; st; st >>= 1) {
    if (t < st) s[t] += s[t + st];
    __syncthreads();
  }
  if (t == 0) classLoss[c0 + g] = s[0];
}

// ---------------- K8: mean over classes ------------------------------------
__global__ void lv_finalize(const float* __restrict__ classLoss, float* __restrict__ out) {
  float v = (threadIdx.x < CDIM) ? classLoss[threadIdx.x] : 0.f;
  #pragma unroll
  for (int d = 16; d; d >>= 1) v += __shfl_down(v, d, 32);
  if (threadIdx.x == 0) out[0] = v / (float)CDIM;
}

// ---------------------------------------------------------------------------
extern "C" void kernel_launch(void* const* d_in, const int* in_sizes, int n_in,
                              void* d_out, int out_size, void* d_ws, size_t ws_size,
                              hipStream_t stream) {
  const float* logits = (const float*)d_in[0];
  const int*   labels = (const int*)d_in[1];
  float*       out    = (float*)d_out;
  char*        ws     = (char*)d_ws;

  unsigned* fgTotU    = (unsigned*)ws;           // 256 B slot
  float*    classLoss = (float*)(ws + 256);      // 256 B slot
  char*     chunkWs   = ws + 512;

  const size_t perClass = (size_t)PDIM * 4 * 2      // keysA + keysB
                        + (size_t)256 * NBLK * 4    // blockHist
                        + (size_t)256 * 4 * 2       // digitTotal + digitBase
                        + (size_t)NBLK * 4 * 2;     // chunkFg + partials
  size_t avail = (ws_size > 512) ? (ws_size - 512) : 0;
  int Gmax = (int)(avail / perClass);
  if (Gmax < 1) Gmax = 1;
  if (Gmax > CDIM) Gmax = CDIM;

  lv_zero_fg<<<1, 32, 0, stream>>>(fgTotU);
  lv_fg_count<<<NBLK, 256, 0, stream>>>(labels, fgTotU);

  for (int c0 = 0; c0 < CDIM; c0 += Gmax) {
    int g = CDIM - c0;
    if (g > Gmax) g = Gmax;

    unsigned* keysA      = (unsigned*)chunkWs;
    unsigned* keysB      = keysA + (size_t)Gmax * PDIM;
    unsigned* blockHist  = keysB + (size_t)Gmax * PDIM;
    unsigned* digitTotal = blockHist + (size_t)Gmax * 256 * NBLK;
    unsigned* digitBase  = digitTotal + (size_t)Gmax * 256;
    unsigned* chunkFg    = digitBase + (size_t)Gmax * 256;
    float*    partials   = (float*)(chunkFg + (size_t)Gmax * NBLK);

    lv_keygen<<<PDIM / 256, 256, 0, stream>>>(logits, labels, keysA, c0, g);

    unsigned* kin  = keysA;
    unsigned* kout = keysB;
    for (int pass = 0; pass < 4; ++pass) {
      int shift = pass * 8;
      lv_hist_pass<<<dim3(NBLK, g), 256, 0, stream>>>(kin, blockHist, shift);
      lv_scan_block_hist<<<g * 64, 128, 0, stream>>>(blockHist, digitTotal, g);
      lv_digit_scan_wmma<<<g, 32, 0, stream>>>(digitTotal, digitBase);
      lv_scatter_pass<<<dim3(NBLK, g), 256, 0, stream>>>(kin, kout, blockHist,
                                                         digitBase, shift);
      unsigned* tmp = kin; kin = kout; kout = tmp;
    }
    // 4 passes -> sorted keys back in keysA (== kin)
    lv_fg_chunk<<<dim3(NBLK, g), 256, 0, stream>>>(kin, chunkFg);
    lv_scan_chunkfg<<<g, 32, 0, stream>>>(chunkFg);
    lv_contrib<<<dim3(NBLK, g), 256, 0, stream>>>(kin, chunkFg, fgTotU, partials, c0);
    lv_class_reduce<<<g, 256, 0, stream>>>(partials, classLoss, c0);
  }
  lv_finalize<<<1, 32, 0, stream>>>(classLoss, out);
}